// OTResample_18279380812196
// MI455X (gfx1250) — compile-verified
//
#include <hip/hip_runtime.h>
#include <hip/hip_bf16.h>

typedef __attribute__((ext_vector_type(2))) float v2f;
typedef __attribute__((ext_vector_type(8))) float v8f;

#define EPS     0.1f
#define NITERS  10
#define BB      8
#define NN      2048
#define DD      64
#define JS      4                       // column-range split for occupancy
// -log(2048) = -11*ln(2)
#define LOG_NU  (-7.6246189861593985f)

__device__ __forceinline__ float fexp(float x) {
    // exp(x) = exp2(x * log2(e)) -> v_exp_f32
    return __builtin_amdgcn_exp2f(x * 1.4426950408889634f);
}
__device__ __forceinline__ float flog(float x) {
    // ln(x) = log2(x) * ln(2) -> v_log_f32
    return __builtin_amdgcn_logf(x) * 0.6931471805599453f;
}

// ---------------------------------------------------------------------------
// Kernel A: per-batch prep.
//   sq[n] = |p_n|^2, logmu[n] = log(w+1e-16)
//   C_mean = 2*(mean(sq) - |mean_p|^2) + 1e-8   (analytic, no NxN pass)
//   a = 1/(C_mean*eps);  f' = g' = -a*sq  (primed potentials, f=g=0)
// ---------------------------------------------------------------------------
__global__ __launch_bounds__(256) void ot_prep(
    const float* __restrict__ part, const float* __restrict__ w,
    float* __restrict__ sq, float* __restrict__ logmu,
    float* __restrict__ fp, float* __restrict__ gp,
    float* __restrict__ a_arr)
{
    const int b = blockIdx.x;
    const int tid = threadIdx.x;
    const float* P = part + (size_t)b * NN * DD;
    __shared__ float red[256];
    __shared__ float col[64];

    float local = 0.f;
    for (int n = tid; n < NN; n += 256) {
        const float* row = P + (size_t)n * DD;
        float s = 0.f;
        #pragma unroll
        for (int d = 0; d < DD; d += 4) {
            float4 v = *reinterpret_cast<const float4*>(row + d);
            s += v.x * v.x + v.y * v.y + v.z * v.z + v.w * v.w;
        }
        sq[b * NN + n] = s;
        local += s;
        logmu[b * NN + n] = flog(w[b * NN + n] + 1e-16f);
    }
    red[tid] = local;
    __syncthreads();
    for (int off = 128; off > 0; off >>= 1) {
        if (tid < off) red[tid] += red[tid + off];
        __syncthreads();
    }
    const float mean_sq = red[0] * (1.f / NN);
    __syncthreads();

    // per-dimension means -> |mean_p|^2
    const int d = tid & 63, g = tid >> 6;
    float pv = 0.f;
    for (int n = g; n < NN; n += 4) pv += P[(size_t)n * DD + d];
    red[tid] = pv;
    __syncthreads();
    if (tid < 64) {
        float v = red[tid] + red[tid + 64] + red[tid + 128] + red[tid + 192];
        v *= (1.f / NN);
        col[tid] = v * v;
    }
    __syncthreads();
    for (int off = 32; off > 0; off >>= 1) {
        if (tid < off) col[tid] += col[tid + off];
        __syncthreads();
    }
    const float c_mean = 2.f * (mean_sq - col[0]) + 1e-8f;
    const float a = 1.f / (c_mean * EPS);
    if (tid == 0) a_arr[b] = a;
    for (int n = tid; n < NN; n += 256) {
        float v = -a * sq[b * NN + n];
        fp[b * NN + n] = v;
        gp[b * NN + n] = v;
    }
}

// ---------------------------------------------------------------------------
// Kernel B1: partial Sinkhorn half-step (one of JS column ranges).
//   (m,s) = online-logsumexp_j( 2a * (p_i . p_j) + vin'_j ),  j in z-slice
// Gram tiles via v_wmma_f32_16x16x4_f32. One v_exp_f32 per tile element.
// Block = 128 threads = 4 waves; each wave owns 16 rows i.
// ---------------------------------------------------------------------------
__global__ __launch_bounds__(128) void ot_sinkhorn_partial(
    const float* __restrict__ part,
    const float* __restrict__ vin,
    const float* __restrict__ a_arr,
    float* __restrict__ pm,            // [JS][BB*NN] partial max
    float* __restrict__ ps)            // [JS][BB*NN] partial sum
{
    const int b    = blockIdx.y;
    const int z    = blockIdx.z;
    const int wave = threadIdx.x >> 5;
    const int lane = threadIdx.x & 31;
    const int half = lane >> 4;
    const int l16  = lane & 15;
    const int i0   = blockIdx.x * 64 + wave * 16;
    const float* P = part + (size_t)b * NN * DD;
    const float s2 = 2.f * a_arr[b];

    // A-fragment (rows i, loop-invariant): lane holds A[M=l16][K=2*half(+1)]
    v2f afrag[16];
    {
        const float* arow = P + (size_t)(i0 + l16) * DD + 2 * half;
        #pragma unroll
        for (int kb = 0; kb < 16; ++kb) {
            afrag[kb].x = arow[4 * kb];
            afrag[kb].y = arow[4 * kb + 1];
        }
    }

    float m[8], s[8];
    #pragma unroll
    for (int r = 0; r < 8; ++r) { m[r] = -__builtin_inff(); s[r] = 0.f; }

    const float* vinb = vin + b * NN;
    const int jbeg = z * (NN / JS);
    const int jend = jbeg + (NN / JS);
    for (int j0 = jbeg; j0 < jend; j0 += 16) {
        v8f c = {0.f, 0.f, 0.f, 0.f, 0.f, 0.f, 0.f, 0.f};
        const float* brow = P + (size_t)(j0 + l16) * DD + 2 * half;
        #pragma unroll
        for (int kb = 0; kb < 16; ++kb) {
            v2f bf;
            bf.x = brow[4 * kb];
            bf.y = brow[4 * kb + 1];
            c = __builtin_amdgcn_wmma_f32_16x16x4_f32(
                    false, afrag[kb], false, bf, (short)0, c, false, false);
        }
        // C/D layout: c[r] is element (i = i0 + r + 8*half, j = j0 + l16)
        const float colv = vinb[j0 + l16];
        #pragma unroll
        for (int r = 0; r < 8; ++r) {
            float v = fmaf(s2, c[r], colv);
            float d = v - m[r];                 // +inf on first tile
            float e = fexp(-__builtin_fabsf(d)); // exp(-|d|); exp(-inf)=0
            s[r] = (d > 0.f) ? fmaf(s[r], e, 1.f) : (s[r] + e);
            m[r] = fmaxf(m[r], v);
        }
    }
    // merge the 16 lane-slices of each row (xor 1,2,4,8 stays inside halves)
    #pragma unroll
    for (int off = 1; off < 16; off <<= 1) {
        #pragma unroll
        for (int r = 0; r < 8; ++r) {
            float om = __shfl_xor(m[r], off, 32);
            float os = __shfl_xor(s[r], off, 32);
            float d  = om - m[r];
            float e  = fexp(-__builtin_fabsf(d));
            s[r] = (d > 0.f) ? fmaf(s[r], e, os) : fmaf(os, e, s[r]);
            m[r] = fmaxf(m[r], om);
        }
    }
    if (l16 == 0) {
        #pragma unroll
        for (int r = 0; r < 8; ++r) {
            const int i = i0 + r + 8 * half;
            pm[(size_t)z * (BB * NN) + b * NN + i] = m[r];
            ps[(size_t)z * (BB * NN) + b * NN + i] = s[r];
        }
    }
}

// ---------------------------------------------------------------------------
// Kernel B2: merge JS partials per row, finish logsumexp, write potential.
//   vout'_i = base_i - ( m + ln(s) )
// ---------------------------------------------------------------------------
__global__ __launch_bounds__(256) void ot_sinkhorn_combine(
    const float* __restrict__ pm, const float* __restrict__ ps,
    const float* __restrict__ logmu, float* __restrict__ vout,
    int use_logmu)
{
    const int idx = blockIdx.x * 256 + threadIdx.x;   // flat over BB*NN
    float m = pm[idx];
    float s = ps[idx];
    #pragma unroll
    for (int z = 1; z < JS; ++z) {
        float om = pm[(size_t)z * (BB * NN) + idx];
        float os = ps[(size_t)z * (BB * NN) + idx];
        float d  = om - m;
        float e  = fexp(-__builtin_fabsf(d));
        s = (d > 0.f) ? fmaf(s, e, os) : fmaf(os, e, s);
        m = fmaxf(m, om);
    }
    const float lse  = m + flog(s);
    const float base = use_logmu ? logmu[idx] : LOG_NU;
    vout[idx] = base - lse;
}

// ---------------------------------------------------------------------------
// Kernel C: output transport.
//   P_nm = exp(f'_n + g'_m + 2a * G_nm);  out[m,d] = N * sum_n P_nm * p[n,d]
// Gram tile via WMMA -> exp -> LDS relayout -> second WMMA chain (P^T @ part).
// Block = 128 threads = 4 waves; each wave owns 16 output rows m.
// ---------------------------------------------------------------------------
__global__ __launch_bounds__(128) void ot_output(
    const float* __restrict__ part,
    const float* __restrict__ fp,
    const float* __restrict__ gp,
    const float* __restrict__ a_arr,
    float* __restrict__ out_part,
    float* __restrict__ out_w)
{
    const int b    = blockIdx.y;
    const int wave = threadIdx.x >> 5;
    const int lane = threadIdx.x & 31;
    const int half = lane >> 4;
    const int l16  = lane & 15;
    const int m0   = blockIdx.x * 64 + wave * 16;
    const float* P = part + (size_t)b * NN * DD;
    const float s2 = 2.f * a_arr[b];

    __shared__ float plds[4][16][16];   // [wave][n_local][m_local], wave-private

    // B-fragment for gram: columns are rows m0..m0+15 (loop-invariant)
    v2f bfragM[16];
    {
        const float* brow = P + (size_t)(m0 + l16) * DD + 2 * half;
        #pragma unroll
        for (int kb = 0; kb < 16; ++kb) {
            bfragM[kb].x = brow[4 * kb];
            bfragM[kb].y = brow[4 * kb + 1];
        }
    }
    const float gpv = gp[b * NN + m0 + l16];

    v8f acc[4];
    #pragma unroll
    for (int dc = 0; dc < 4; ++dc)
        acc[dc] = (v8f){0.f, 0.f, 0.f, 0.f, 0.f, 0.f, 0.f, 0.f};

    for (int n0 = 0; n0 < NN; n0 += 16) {
        // gram tile G[n(16) x m(16)]
        v8f c = {0.f, 0.f, 0.f, 0.f, 0.f, 0.f, 0.f, 0.f};
        const float* arow = P + (size_t)(n0 + l16) * DD + 2 * half;
        #pragma unroll
        for (int kb = 0; kb < 16; ++kb) {
            v2f af;
            af.x = arow[4 * kb];
            af.y = arow[4 * kb + 1];
            c = __builtin_amdgcn_wmma_f32_16x16x4_f32(
                    false, af, false, bfragM[kb], (short)0, c, false, false);
        }
        // P tile -> LDS (c[r] is element (n = n0 + r + 8*half, m = m0 + l16))
        #pragma unroll
        for (int r = 0; r < 8; ++r) {
            const int n = n0 + r + 8 * half;
            plds[wave][r + 8 * half][l16] =
                fexp(fp[b * NN + n] + gpv + s2 * c[r]);
        }
        // acc[m,d] += P^T[m x n] @ part[n x d], K=4 steps over n
        #pragma unroll
        for (int kb2 = 0; kb2 < 4; ++kb2) {
            v2f af;  // A[M=m=l16][K = kb2*4 + 2*half (+1)] = P[n_local][m]
            af.x = plds[wave][kb2 * 4 + 2 * half][l16];
            af.y = plds[wave][kb2 * 4 + 2 * half + 1][l16];
            const float* prow = P + (size_t)(n0 + kb2 * 4 + 2 * half) * DD;
            #pragma unroll
            for (int dc = 0; dc < 4; ++dc) {
                v2f bf;  // B[K][N=d=dc*16+l16]
                bf.x = prow[dc * 16 + l16];
                bf.y = prow[DD + dc * 16 + l16];
                acc[dc] = __builtin_amdgcn_wmma_f32_16x16x4_f32(
                              false, af, false, bf, (short)0, acc[dc], false, false);
            }
        }
    }
    // D layout: acc[dc][r] is element (m = m0 + r + 8*half, d = dc*16 + l16)
    #pragma unroll
    for (int dc = 0; dc < 4; ++dc) {
        #pragma unroll
        for (int r = 0; r < 8; ++r) {
            const int mrow = m0 + r + 8 * half;
            out_part[((size_t)b * NN + mrow) * DD + dc * 16 + l16] =
                (float)NN * acc[dc][r];
        }
    }
    if (threadIdx.x < 64)
        out_w[b * NN + blockIdx.x * 64 + threadIdx.x] = 1.f / NN;
}

// ---------------------------------------------------------------------------
extern "C" void kernel_launch(void* const* d_in, const int* in_sizes, int n_in,
                              void* d_out, int out_size, void* d_ws, size_t ws_size,
                              hipStream_t stream)
{
    const float* part = (const float*)d_in[0];   // [8, 2048, 64] fp32
    const float* w    = (const float*)d_in[1];   // [8, 2048]     fp32

    float* out_part = (float*)d_out;                         // [8,2048,64]
    float* out_w    = out_part + (size_t)BB * NN * DD;       // [8,2048]

    float* ws    = (float*)d_ws;           // ~786 KB total
    float* sq    = ws;                     // B*N
    float* logmu = sq + BB * NN;           // B*N
    float* fp    = logmu + BB * NN;        // B*N  (f' = f - a*sq)
    float* gp    = fp + BB * NN;           // B*N  (g' = g - a*sq)
    float* a_arr = gp + BB * NN;           // B
    float* pm    = a_arr + BB;             // JS*B*N partial max
    float* psum  = pm + (size_t)JS * BB * NN; // JS*B*N partial sum

    ot_prep<<<BB, 256, 0, stream>>>(part, w, sq, logmu, fp, gp, a_arr);

    const dim3 gpart(NN / 64, BB, JS);
    const int  gcomb = (BB * NN) / 256;
    for (int it = 0; it < NITERS; ++it) {
        // f-update reads g', writes f' (base = log_mu)
        ot_sinkhorn_partial<<<gpart, 128, 0, stream>>>(part, gp, a_arr, pm, psum);
        ot_sinkhorn_combine<<<gcomb, 256, 0, stream>>>(pm, psum, logmu, fp, 1);
        // g-update reads fresh f', writes g' (base = -log N)
        ot_sinkhorn_partial<<<gpart, 128, 0, stream>>>(part, fp, a_arr, pm, psum);
        ot_sinkhorn_combine<<<gcomb, 256, 0, stream>>>(pm, psum, logmu, gp, 0);
    }

    ot_output<<<dim3(NN / 64, BB), 128, 0, stream>>>(
        part, fp, gp, a_arr, out_part, out_w);
}